// DifferentiableSynth_84095459655724
// MI455X (gfx1250) — compile-verified
//
#include <hip/hip_runtime.h>
#include <math.h>

typedef __attribute__((ext_vector_type(2))) float v2f;
typedef __attribute__((ext_vector_type(8))) float v8f;
typedef __attribute__((ext_vector_type(4))) unsigned int u32x4;
typedef __attribute__((ext_vector_type(4))) int i32x4;
typedef __attribute__((ext_vector_type(8))) int i32x8;
typedef int v4i __attribute__((vector_size(16)));   // matches builtin param type

#define AS1 __attribute__((address_space(1)))
#define AS3 __attribute__((address_space(3)))

#define N_SAMPLES 14400000      // 48000 * 300
#define BLK       1024
#define NBLK      14063         // ceil(N/1024); PAD = 512
#define SRf       48000.0f
#define TWO_PI    6.283185307179586

// ---- gfx1250 data-mover availability (device pass only; host pass sees fallbacks) ----
#if defined(__HIP_DEVICE_COMPILE__) && __has_builtin(__builtin_amdgcn_tensor_load_to_lds)
#define HAVE_TDM 1
#else
#define HAVE_TDM 0
#endif
#if defined(__HIP_DEVICE_COMPILE__) && __has_builtin(__builtin_amdgcn_global_load_async_to_lds_b128)
#define HAVE_ASYNC_LD 1
#else
#define HAVE_ASYNC_LD 0
#endif
#if defined(__HIP_DEVICE_COMPILE__) && __has_builtin(__builtin_amdgcn_global_store_async_from_lds_b128)
#define HAVE_ASYNC_ST 1
#else
#define HAVE_ASYNC_ST 0
#endif

struct Params {
    float noise_mix;
    float amp_a, amp_d, amp_s, amp_r;
    float cutoff_base, filter_q, env_amount;
    float fe_a, fe_d, fe_s, fe_r;
    double w, ef, dsf, k1, inv1mr;
};

__device__ __forceinline__ float sigm(float x) { return 1.0f / (1.0f + __expf(-x)); }

__device__ __forceinline__ float adsr_env(float t, float A, float D, float S, float R) {
    float a = A * SRf, d = D * SRf, r = R * SRf;
    float ae = a, de = a + d;
    float se = fmaxf(de, (float)N_SAMPLES - r);
    float env;
    if (t < ae)       env = t / fmaxf(a, 1e-5f);
    else if (t >= se) env = S * (1.0f - (t - se) / fmaxf(r, 1e-5f));
    else if (t < de)  env = 1.0f - (t - ae) / fmaxf(d, 1e-5f) * (1.0f - S);
    else              env = S;
    return fminf(fmaxf(env, 0.0f), 1.0f);
}

__device__ __forceinline__ void wait_async0() {
#if defined(__HIP_DEVICE_COMPILE__)
#if __has_builtin(__builtin_amdgcn_s_wait_asynccnt)
    __builtin_amdgcn_s_wait_asynccnt(0);
#else
    asm volatile("s_wait_asynccnt 0x0" ::: "memory");
#endif
#endif
}

#if HAVE_TDM
// One-row TDM tile: 1024 f32 elements, tensor_dim0 = valid elements so the
// TDM's OOB-returns-zero semantics zero-fill the padded tail for free.
__device__ __forceinline__ void tdm_load_row(unsigned lds_addr, const AS1 float* gptr,
                                             unsigned valid_elems) {
    unsigned long long ga = (unsigned long long)gptr;
    u32x4 g0;
    g0.x = 1u;                                              // count=1, user descriptor
    g0.y = lds_addr;                                        // lds_addr -> bits [63:32]
    g0.z = (unsigned)(ga & 0xFFFFFFFFu);                    // global_addr[31:0]
    g0.w = (unsigned)((ga >> 32) & 0x01FFFFFFu) | (2u << 30); // ga[56:32] | type=2
    i32x8 g1;
    g1[0] = (int)(2u << 16);                                // data_size = 4B, mask=0
    g1[1] = (int)((valid_elems & 0xFFFFu) << 16);           // tensor_dim0[15:0]
    g1[2] = (int)((valid_elems >> 16) & 0xFFFFu);           // tensor_dim0[31:16]
    g1[3] = (int)(1024u << 16);                             // tile_dim0 = 1024
    g1[4] = 0;                                              // tile_dim1/2 unused (1D)
    g1[5] = 1024;                                           // dim0 stride (unused, 1 row)
    g1[6] = 0;
    g1[7] = 0;
    i32x4 z4 = {0, 0, 0, 0};
#if __clang_major__ >= 23
    i32x8 z8 = {0, 0, 0, 0, 0, 0, 0, 0};
    __builtin_amdgcn_tensor_load_to_lds(g0, g1, z4, z4, z8, 0);
#else
    __builtin_amdgcn_tensor_load_to_lds(g0, g1, z4, z4, 0);
#endif
}
#endif

// ---------------- Kernel A: fold 15 raw scalars into derived params ----------------
__global__ void k_params(const float* nm, const float* sf, const float* ef,
                         const float* pd, const float* aa, const float* ad,
                         const float* as_, const float* ar, const float* fc,
                         const float* fq, const float* fea, const float* fa,
                         const float* fd, const float* fs, const float* fr,
                         Params* P)
{
    if (threadIdx.x == 0 && blockIdx.x == 0) {
        P->noise_mix = sigm(nm[0]);
        float start_freq = 20.0f + sigm(sf[0]) * 7980.0f;
        float end_freq   = 20.0f + sigm(ef[0]) * 7980.0f;
        float pitch_dec  = 0.01f + sigm(pd[0]) * 1.99f;
        P->amp_a = 0.001f + sigm(aa[0]) * 0.999f;
        P->amp_d = 0.01f  + sigm(ad[0]) * 1.99f;
        P->amp_s = sigm(as_[0]);
        P->amp_r = 0.01f  + sigm(ar[0]) * 1.99f;
        P->cutoff_base = 100.0f + sigm(fc[0]) * 11900.0f;
        P->filter_q    = 0.707f + sigm(fq[0]) * 9.293f;
        P->env_amount  = tanhf(fea[0]) * 8000.0f;
        P->fe_a = 0.001f + sigm(fa[0]) * 0.999f;
        P->fe_d = 0.01f  + sigm(fd[0]) * 0.99f;
        P->fe_s = sigm(fs[0]);
        P->fe_r = 0.01f  + sigm(fr[0]) * 0.99f;
        // closed-form phase constants: cumsum(freq) = (i+1)*ef + (sf-ef)*(1-r^(i+1))/(1-r)
        double dt = 300.0 / (double)(N_SAMPLES - 1);
        double k1 = dt / ((double)pitch_dec + 1e-6);
        P->k1     = k1;
        P->inv1mr = 1.0 / (-expm1(-k1));     // 1/(1-r), r = exp(-k1)
        P->w      = TWO_PI / 48000.0;
        P->ef     = (double)end_freq;
        P->dsf    = (double)start_freq - (double)end_freq;
    }
}

// ---------------- Kernel B: per-block cutoff means via WMMA ones-reduction ----------------
__global__ __launch_bounds__(32) void k_coeffs(const Params* __restrict__ P,
                                               float* __restrict__ coeffs)
{
    __shared__ float Pl[32 * 16];           // partials: [src lane][block]
    const int lane = threadIdx.x;
    const int blk0 = blockIdx.x * 16;
    const float cb = P->cutoff_base, ea = P->env_amount;
    const float fa = P->fe_a, fd = P->fe_d, fs = P->fe_s, fr = P->fe_r;

    float part[16];
    #pragma unroll
    for (int b = 0; b < 16; ++b) {
        float s = 0.0f;
        int base = (blk0 + b) * BLK + lane * 32;
        for (int k = 0; k < 32; ++k) {
            int idx = base + k;
            float v = 0.0f;
            if (idx < N_SAMPLES) {
                float fe = adsr_env((float)idx, fa, fd, fs, fr);
                float c  = cb + fe * ea;
                v = fminf(fmaxf(c, 20.0f), 48000.0f / 2.1f);
            }
            s += v;
        }
        part[b] = s;
    }
    #pragma unroll
    for (int b = 0; b < 16; ++b) Pl[lane * 16 + b] = part[b];
    __syncthreads();

    // D = ones(16x32) * P(32x16) via 8 chained exact-f32 WMMAs (K=4 each).
    v8f acc = {0.f, 0.f, 0.f, 0.f, 0.f, 0.f, 0.f, 0.f};
    const v2f ones = {1.0f, 1.0f};
    const int col  = lane & 15;
    const int hsel = (lane >> 4) << 1;
    #pragma unroll
    for (int c = 0; c < 8; ++c) {
        v2f bf;
        bf.x = Pl[(4 * c + hsel + 0) * 16 + col];
        bf.y = Pl[(4 * c + hsel + 1) * 16 + col];
        acc = __builtin_amdgcn_wmma_f32_16x16x4_f32(
            false, ones, false, bf, (short)0, acc, false, false);
    }

    if (lane < 16) {
        int blk = blk0 + lane;
        if (blk < NBLK) {
            float mean  = acc[0] * (1.0f / 1024.0f);
            float w0    = (float)TWO_PI * mean / SRf;
            float sw    = sinf(w0), cw = cosf(w0);
            float alpha = sw / (2.0f * P->filter_q);
            float b0    = (1.0f - cw) * 0.5f;
            float b1    = 1.0f - cw;
            float inv   = 1.0f / (1.0f + alpha + 1e-8f);
            float* o = coeffs + blk * 5;
            o[0] = b0 * inv;
            o[1] = b1 * inv;
            o[2] = b0 * inv;
            o[3] = (-2.0f * cw) * inv;
            o[4] = (1.0f - alpha) * inv;
        }
    }
}

// ---------------- Kernel C: TDM-staged noise + synthesis + IIR + async store ----------------
__global__ __launch_bounds__(32) void k_synth(const Params* __restrict__ P,
                                              const float* __restrict__ coeffs,
                                              const float* __restrict__ noise,
                                              float* __restrict__ out)
{
    __shared__ float sb[BLK];   // sine -> source -> y -> y*env
    __shared__ float nb[BLK];   // noise staged by TDM / async DMA
    const int j    = blockIdx.x;
    const int lane = threadIdx.x;
    const int base = j * BLK;
    const float  nm = P->noise_mix;
    const double w  = P->w,  ef = P->ef, dsf = P->dsf;
    const double k1 = P->k1, inv1mr = P->inv1mr;

    // ---- stage this block's noise into LDS (overlapped with sine computation) ----
#if HAVE_TDM
    {
        unsigned lds_addr = (unsigned)(unsigned long long)(AS3 float*)nb;
        const AS1 float* g = (const AS1 float*)noise + base;
        tdm_load_row(lds_addr, g, (unsigned)(N_SAMPLES - base));
    }
#elif HAVE_ASYNC_LD
    {
        AS3 v4i* l = (AS3 v4i*)(AS3 float*)nb;
        AS1 v4i* g = (AS1 v4i*)((AS1 float*)(float*)noise + base);
        #pragma unroll
        for (int c = 0; c < 8; ++c) {
            if (base + (c + 1) * 128 <= N_SAMPLES) {     // N % 128 == 0: chunks all-or-nothing
                __builtin_amdgcn_global_load_async_to_lds_b128(g + c * 32 + lane,
                                                               l + c * 32 + lane, 0, 0);
            }
        }
    }
#else
    for (int c = 0; c < 32; ++c) {
        int ib = c * 32 + lane, i = base + ib;
        nb[ib] = (i < N_SAMPLES) ? noise[i] : 0.0f;
    }
#endif

    // ---- closed-form phase -> sine (f64 prefix-sum closed form, no global scan) ----
    for (int c = 0; c < 32; ++c) {
        int ib = c * 32 + lane, i = base + ib;
        float v = 0.0f;
        if (i < N_SAMPLES) {
            double ip1 = (double)(i + 1);
            double cum = ip1 * ef + dsf * (-expm1(-ip1 * k1)) * inv1mr;
            double ph  = fmod(w * cum, TWO_PI);
            v = sinf((float)ph);
        }
        sb[ib] = v;
    }

    // ---- wait for the noise DMA, then mix source = (1-nm)*sine + nm*noise ----
#if HAVE_TDM
    __builtin_amdgcn_s_wait_tensorcnt(0);
#elif HAVE_ASYNC_LD
    wait_async0();
#endif
    __syncthreads();
    for (int c = 0; c < 32; ++c) {
        int ib = c * 32 + lane, i = base + ib;
        if (i < N_SAMPLES) sb[ib] = (1.0f - nm) * sb[ib] + nm * nb[ib];
    }
    __syncthreads();

    // ---- serial biquad over the block (zero initial state, per reference scan) ----
    if (lane == 0) {
        const float* co = coeffs + j * 5;
        float b0 = co[0], b1 = co[1], b2 = co[2], a1 = co[3], a2 = co[4];
        float x1 = 0.f, x2 = 0.f, y1 = 0.f, y2 = 0.f;
        for (int t = 0; t < BLK; ++t) {
            float x   = sb[t];
            float fir = b0 * x + b1 * x1 + b2 * x2;
            float y   = fir - a1 * y1 - a2 * y2;
            sb[t] = y;
            x2 = x1; x1 = x;
            y2 = y1; y1 = y;
        }
    }
    __syncthreads();

    // ---- amplitude envelope, then async store LDS -> global ----
    const float Aa = P->amp_a, Ad = P->amp_d, As = P->amp_s, Ar = P->amp_r;
    for (int c = 0; c < 32; ++c) {
        int ib = c * 32 + lane, i = base + ib;
        if (i < N_SAMPLES) sb[ib] *= adsr_env((float)i, Aa, Ad, As, Ar);
    }
    __syncthreads();

#if HAVE_ASYNC_ST
    {
        AS3 v4i* l = (AS3 v4i*)(AS3 float*)sb;
        AS1 v4i* g = (AS1 v4i*)((AS1 float*)(float*)out + base);
        #pragma unroll
        for (int c = 0; c < 8; ++c) {
            if (base + (c + 1) * 128 <= N_SAMPLES) {     // N % 128 == 0: chunks all-or-nothing
                __builtin_amdgcn_global_store_async_from_lds_b128(g + c * 32 + lane,
                                                                  l + c * 32 + lane, 0, 0);
            }
        }
        wait_async0();
    }
#else
    for (int c = 0; c < 32; ++c) {
        int ib = c * 32 + lane, i = base + ib;
        if (i < N_SAMPLES) out[i] = sb[ib];
    }
#endif
}

extern "C" void kernel_launch(void* const* d_in, const int* in_sizes, int n_in,
                              void* d_out, int out_size, void* d_ws, size_t ws_size,
                              hipStream_t stream)
{
    (void)in_sizes; (void)n_in; (void)out_size; (void)ws_size;
    const float* in[16];
    for (int i = 0; i < 16; ++i) in[i] = (const float*)d_in[i];
    Params* P      = (Params*)d_ws;
    float*  coeffs = (float*)((char*)d_ws + 256);   // NBLK*5*4 ≈ 281 KB scratch
    float*  out    = (float*)d_out;
    const float* noise = in[15];

    k_params<<<1, 1, 0, stream>>>(in[0], in[1], in[2], in[3], in[4], in[5], in[6],
                                  in[7], in[8], in[9], in[10], in[11], in[12],
                                  in[13], in[14], P);
    k_coeffs<<<(NBLK + 15) / 16, 32, 0, stream>>>(P, coeffs);
    k_synth<<<NBLK, 32, 0, stream>>>(P, coeffs, noise, out);
}